// PointNetFeaturePropagation_43087111913599
// MI455X (gfx1250) — compile-verified
//
#include <hip/hip_runtime.h>

typedef __attribute__((ext_vector_type(2))) float v2f;
typedef __attribute__((ext_vector_type(8))) float v8f;

#define BATCH 4
#define NPTS  16384
#define SPTS  2048
#define C1    128
#define C2    256
#define HID   256
#define K1    (C1 + C2)   /* 384 */
#define BN_EPS 1e-5f

#define AS_STRIDE 36      /* 32 used + pad: conflict-free A-frag b64 reads */
#define HS_STRIDE 258     /* 256 used + pad: conflict-free phase-B A-frags */
#define SMEM_FLOATS (32 * 256 + 64 * HS_STRIDE)   /* Bs + Hs(aliases As) */
#define SMEM_BYTES  (SMEM_FLOATS * 4)             /* 98816 B */

// ---------------------------------------------------------------------------
// Kernel 1: fused 3-NN search + inverse-distance interpolation + concat.
// Produces xbuf[b*N+n][0:128] = points1 row, [128:384] = interpolated row.
// ---------------------------------------------------------------------------
__global__ __launch_bounds__(256) void knn_interp_kernel(
    const float* __restrict__ xyz1, const float* __restrict__ xyz2,
    const float* __restrict__ points1, const float* __restrict__ points2,
    float* __restrict__ xbuf)
{
  __shared__ float4 p2s[SPTS];      // xyz + |p|^2 of reference points (32 KB)
  __shared__ int    idxs[256][3];
  __shared__ float  wgts[256][3];

  const int tid    = threadIdx.x;
  const int b      = blockIdx.y;
  const int nblock = blockIdx.x;

  for (int s = tid; s < SPTS; s += 256) {
    const float* p = xyz2 + (b * SPTS + s) * 3;
    float x = p[0], y = p[1], z = p[2];
    p2s[s] = make_float4(x, y, z, x * x + y * y + z * z);
  }
  __syncthreads();

  const int n = nblock * 256 + tid;
  const float* q = xyz1 + (b * NPTS + n) * 3;
  const float qx = q[0], qy = q[1], qz = q[2];
  const float qn = qx * qx + qy * qy + qz * qz;

  // top-3 smallest squared distances (strict < keeps lowest index on ties,
  // matching jax.lax.top_k stability)
  float d0 = 1e30f, d1 = 1e30f, d2 = 1e30f;
  int   i0 = 0,     i1 = 0,     i2 = 0;
  for (int s = 0; s < SPTS; ++s) {
    float4 p = p2s[s];
    float d = qn + p.w - 2.0f * (qx * p.x + qy * p.y + qz * p.z);
    if (d < d0)      { d2 = d1; i2 = i1; d1 = d0; i1 = i0; d0 = d; i0 = s; }
    else if (d < d1) { d2 = d1; i2 = i1; d1 = d;  i1 = s; }
    else if (d < d2) { d2 = d;  i2 = s; }
  }

  float e0 = fmaxf(sqrtf(fmaxf(d0, 0.0f)), 1e-10f);
  float e1 = fmaxf(sqrtf(fmaxf(d1, 0.0f)), 1e-10f);
  float e2 = fmaxf(sqrtf(fmaxf(d2, 0.0f)), 1e-10f);
  float w0 = 1.0f / e0, w1 = 1.0f / e1, w2 = 1.0f / e2;
  float ws = 1.0f / (w0 + w1 + w2);
  idxs[tid][0] = i0;      idxs[tid][1] = i1;      idxs[tid][2] = i2;
  wgts[tid][0] = w0 * ws; wgts[tid][1] = w1 * ws; wgts[tid][2] = w2 * ws;
  __syncthreads();

  // cooperative, coalesced assembly of the concat buffer
  for (int p = 0; p < 256; ++p) {
    const int   j0 = idxs[p][0], j1 = idxs[p][1], j2 = idxs[p][2];
    const float a0 = wgts[p][0], a1 = wgts[p][1], a2 = wgts[p][2];
    const int gp = b * NPTS + nblock * 256 + p;
    float*       dst = xbuf + gp * K1;
    const float* f1  = points1 + gp * C1;
    const float* r0  = points2 + (b * SPTS + j0) * C2;
    const float* r1  = points2 + (b * SPTS + j1) * C2;
    const float* r2  = points2 + (b * SPTS + j2) * C2;
    for (int c = tid; c < K1; c += 256) {
      float v;
      if (c < C1) {
        v = f1[c];
      } else {
        const int cc = c - C1;
        v = a0 * r0[cc] + a1 * r1[cc] + a2 * r2[cc];
      }
      dst[c] = v;
    }
  }
}

// ---------------------------------------------------------------------------
// Kernel 2: W [HID x K] -> Wt [K x HID] so GEMM B-tiles load coalesced.
// ---------------------------------------------------------------------------
__global__ __launch_bounds__(256) void transpose_w_kernel(
    const float* __restrict__ W, float* __restrict__ Wt, int K)
{
  const int idx = blockIdx.x * 256 + threadIdx.x;  // grid = K blocks
  const int k = idx >> 8;
  const int n = idx & 255;
  Wt[idx] = W[n * K + k];
}

// ---------------------------------------------------------------------------
// Kernel 3: fused 2-layer MLP.  out = relu(bn2( relu(bn1(x@W1t)) @ W2t )).
// Block = 64 rows x all 256 cols; layer-1 activations never leave LDS.
// 8 waves, each owns 64x32 slab = 4(M) x 2(N) f32 WMMA accumulators.
// Register-prefetch double buffering on the global->LDS staging.
// ---------------------------------------------------------------------------
__global__ __launch_bounds__(256) void mlp2_fused_kernel(
    const float* __restrict__ A,   const float* __restrict__ Wt1,
    const float* __restrict__ Wt2,
    const float* __restrict__ g1, const float* __restrict__ be1,
    const float* __restrict__ mu1, const float* __restrict__ va1,
    const float* __restrict__ g2, const float* __restrict__ be2,
    const float* __restrict__ mu2, const float* __restrict__ va2,
    float* __restrict__ out)
{
  extern __shared__ float smem[];
  float* Bs = smem;                 // [32][256] weight stage (both layers)
  float* Hs = smem + 32 * 256;      // [64][HS_STRIDE] layer-1 activations
  float* As = Hs;                   // [64][AS_STRIDE] aliases Hs (phase A only)

  const int tid   = threadIdx.x;
  const int wave  = tid >> 5;
  const int lane  = tid & 31;
  const int half  = lane >> 4;      // 0: lanes 0-15, 1: lanes 16-31
  const int l16   = lane & 15;
  const int m0    = blockIdx.x * 64;
  const int nbase = wave * 32;      // each wave covers 32 output columns

  float4 aReg[2];
  float4 bReg[8];

  auto loadA = [&](int k0) {
#pragma unroll
    for (int i = 0; i < 2; ++i) {
      const int e = tid + i * 256;          // 512 float4 = 64 rows x 8
      aReg[i] = *(const float4*)(A + (m0 + (e >> 3)) * K1 + k0 + (e & 7) * 4);
    }
  };
  auto storeA = [&]() {
#pragma unroll
    for (int i = 0; i < 2; ++i) {
      const int e = tid + i * 256;
      *(float4*)&As[(e >> 3) * AS_STRIDE + (e & 7) * 4] = aReg[i];
    }
  };
  auto loadB = [&](const float* Bt, int k0) {
#pragma unroll
    for (int i = 0; i < 8; ++i) {
      const int e = tid + i * 256;          // 2048 float4 = 32 rows x 64
      bReg[i] = *(const float4*)(Bt + (k0 + (e >> 6)) * 256 + (e & 63) * 4);
    }
  };
  auto storeB = [&]() {
#pragma unroll
    for (int i = 0; i < 8; ++i) {
      const int e = tid + i * 256;
      *(float4*)&Bs[(e >> 6) * 256 + (e & 63) * 4] = bReg[i];
    }
  };

  // ================= Phase A: h = relu(bn1(x @ W1t)) =================
  v8f acc[4][2];
#pragma unroll
  for (int mt = 0; mt < 4; ++mt)
#pragma unroll
    for (int nt = 0; nt < 2; ++nt)
      acc[mt][nt] = (v8f){0.f, 0.f, 0.f, 0.f, 0.f, 0.f, 0.f, 0.f};

  loadA(0);
  loadB(Wt1, 0);
  for (int k0 = 0; k0 < K1; k0 += 32) {
    storeA();
    storeB();
    __syncthreads();
    if (k0 + 32 < K1) {               // prefetch next tiles (overlaps WMMAs)
      loadA(k0 + 32);
      loadB(Wt1, k0 + 32);
    }
    for (int kk = 0; kk < 32; kk += 4) {
      const int kb = kk + half * 2;   // VGPR0 holds K{0,2}, VGPR1 holds K{1,3}
      v2f afrag[4];
#pragma unroll
      for (int mt = 0; mt < 4; ++mt)
        afrag[mt] = *(const v2f*)&As[(mt * 16 + l16) * AS_STRIDE + kb];
      v2f bfrag[2];
#pragma unroll
      for (int nt = 0; nt < 2; ++nt) {
        const int nn = nbase + nt * 16 + l16;
        v2f bv;
        bv.x = Bs[kb * 256 + nn];
        bv.y = Bs[(kb + 1) * 256 + nn];
        bfrag[nt] = bv;
      }
#pragma unroll
      for (int mt = 0; mt < 4; ++mt)
#pragma unroll
        for (int nt = 0; nt < 2; ++nt)
          acc[mt][nt] = __builtin_amdgcn_wmma_f32_16x16x4_f32(
              false, afrag[mt], false, bfrag[nt],
              (short)0, acc[mt][nt], false, false);
    }
    __syncthreads();                  // retires all As/Bs reads
  }

  // epilogue 1: bn1 + relu -> Hs (overwrites the As alias; safe post-barrier)
#pragma unroll
  for (int nt = 0; nt < 2; ++nt) {
    const int nn = nbase + nt * 16 + l16;
    const float s   = g1[nn] * rsqrtf(va1[nn] + BN_EPS);
    const float off = be1[nn] - mu1[nn] * s;
#pragma unroll
    for (int mt = 0; mt < 4; ++mt)
#pragma unroll
      for (int r = 0; r < 8; ++r) {
        const int ml = mt * 16 + r + half * 8;
        Hs[ml * HS_STRIDE + nn] = fmaxf(acc[mt][nt][r] * s + off, 0.0f);
      }
  }

  // ================= Phase B: out = relu(bn2(h @ W2t)) =================
  v8f acc2[4][2];
#pragma unroll
  for (int mt = 0; mt < 4; ++mt)
#pragma unroll
    for (int nt = 0; nt < 2; ++nt)
      acc2[mt][nt] = (v8f){0.f, 0.f, 0.f, 0.f, 0.f, 0.f, 0.f, 0.f};

  loadB(Wt2, 0);
  for (int k0 = 0; k0 < HID; k0 += 32) {
    storeB();
    __syncthreads();                  // also publishes Hs on first iteration
    if (k0 + 32 < HID) loadB(Wt2, k0 + 32);
    for (int kk = 0; kk < 32; kk += 4) {
      const int kb = kk + half * 2;
      v2f afrag[4];
#pragma unroll
      for (int mt = 0; mt < 4; ++mt)
        afrag[mt] = *(const v2f*)&Hs[(mt * 16 + l16) * HS_STRIDE + k0 + kb];
      v2f bfrag[2];
#pragma unroll
      for (int nt = 0; nt < 2; ++nt) {
        const int nn = nbase + nt * 16 + l16;
        v2f bv;
        bv.x = Bs[kb * 256 + nn];
        bv.y = Bs[(kb + 1) * 256 + nn];
        bfrag[nt] = bv;
      }
#pragma unroll
      for (int mt = 0; mt < 4; ++mt)
#pragma unroll
        for (int nt = 0; nt < 2; ++nt)
          acc2[mt][nt] = __builtin_amdgcn_wmma_f32_16x16x4_f32(
              false, afrag[mt], false, bfrag[nt],
              (short)0, acc2[mt][nt], false, false);
    }
    __syncthreads();
  }

  // epilogue 2: bn2 + relu -> global output
#pragma unroll
  for (int nt = 0; nt < 2; ++nt) {
    const int nn = nbase + nt * 16 + l16;
    const float s   = g2[nn] * rsqrtf(va2[nn] + BN_EPS);
    const float off = be2[nn] - mu2[nn] * s;
#pragma unroll
    for (int mt = 0; mt < 4; ++mt)
#pragma unroll
      for (int r = 0; r < 8; ++r) {
        const int m = m0 + mt * 16 + r + half * 8;
        out[m * HID + nn] = fmaxf(acc2[mt][nt][r] * s + off, 0.0f);
      }
  }
}

// ---------------------------------------------------------------------------
extern "C" void kernel_launch(void* const* d_in, const int* in_sizes, int n_in,
                              void* d_out, int out_size, void* d_ws, size_t ws_size,
                              hipStream_t stream) {
  const float* xyz1    = (const float*)d_in[0];
  const float* xyz2    = (const float*)d_in[1];
  const float* points1 = (const float*)d_in[2];
  const float* points2 = (const float*)d_in[3];
  const float* W1      = (const float*)d_in[4];
  const float* gamma1  = (const float*)d_in[5];
  const float* beta1   = (const float*)d_in[6];
  const float* rm1     = (const float*)d_in[7];
  const float* rv1     = (const float*)d_in[8];
  const float* W2      = (const float*)d_in[9];
  const float* gamma2  = (const float*)d_in[10];
  const float* beta2   = (const float*)d_in[11];
  const float* rm2     = (const float*)d_in[12];
  const float* rv2     = (const float*)d_in[13];
  float* outp = (float*)d_out;

  const int M = BATCH * NPTS;               // 65536 rows
  float* xbuf = (float*)d_ws;               // [M x 384]
  float* Wt1  = xbuf + (size_t)M * K1;      // [384 x 256]
  float* Wt2  = Wt1 + (size_t)K1 * HID;     // [256 x 256]

  transpose_w_kernel<<<K1, 256, 0, stream>>>(W1, Wt1, K1);
  transpose_w_kernel<<<HID, 256, 0, stream>>>(W2, Wt2, HID);

  knn_interp_kernel<<<dim3(NPTS / 256, BATCH), 256, 0, stream>>>(
      xyz1, xyz2, points1, points2, xbuf);

  mlp2_fused_kernel<<<M / 64, 256, SMEM_BYTES, stream>>>(
      xbuf, Wt1, Wt2,
      gamma1, beta1, rm1, rv1,
      gamma2, beta2, rm2, rv2,
      outp);
}